// AlgelogicNetwork_90108413870080
// MI455X (gfx1250) — compile-verified
//
#include <hip/hip_runtime.h>
#include <math.h>

typedef float v2f  __attribute__((ext_vector_type(2)));
typedef float v8f  __attribute__((ext_vector_type(8)));

// Problem constants from the reference
#define MM 16
#define JJ 2
#define II 3
#define LL 2
#define WW 9

#define LOG2E 1.44269504088896340736f

__global__ __launch_bounds__(256) void algelogic_wmma_kernel(
    const float* __restrict__ state,      // (B, 18)
    const float* __restrict__ constants,  // (16, 3, 2)
    const float* __restrict__ gammas,     // (16, 3, 2)
    const float* __restrict__ head_W,     // (16, 2, 3, 2)
    const float* __restrict__ head_b,     // (16, 2, 3)
    const float* __restrict__ tail_W,     // (16, 2, 3)
    const float* __restrict__ tail_b,     // (16, 2)
    float* __restrict__ out,              // (B, 9)
    int Bsz)
{
    const int lane = threadIdx.x & 31;
    const int m    = lane & 15;          // lanes 16..31 duplicate m of lanes 0..15
    const bool hi  = lane >= 16;
    const int wcol = lane & 15;          // WMMA output column handled by this lane
    const bool colvalid = wcol < WW;
    const float fone = colvalid ? 1.0f : 0.0f;   // K=2 feature (constant term), loop-invariant
    const int wclamp = colvalid ? wcol : 0;      // safe column index for feature load

    // ------------------------------------------------------------------
    // Per-m, loop-invariant parameter folding (done once per thread).
    // ------------------------------------------------------------------
    float g[JJ][LL], omg[JJ][LL], cst[JJ][LL];
    #pragma unroll
    for (int j = 0; j < JJ; ++j)
        #pragma unroll
        for (int l = 0; l < LL; ++l) {
            float ga = gammas[m*6 + j*2 + l];
            float sg = 1.0f / (1.0f + __expf(-ga));
            g[j][l]   = sg;
            omg[j][l] = 1.0f - sg;
            cst[j][l] = constants[m*6 + j*2 + l];
        }

    // total_match[w] = K0 + P0*x + Q0*x^2 + P1*y + Q1*y^2
    float K0 = 0.0f, P[LL], Q[LL];
    #pragma unroll
    for (int l = 0; l < LL; ++l) {
        float p = 0.0f, q = 0.0f;
        #pragma unroll
        for (int j = 0; j < JJ; ++j) {
            q += omg[j][l];
            p += omg[j][l] * cst[j][l];
            K0 += omg[j][l] * cst[j][l] * cst[j][l];
        }
        Q[l] = q;
        P[l] = -2.0f * p;
    }

    // captured[i] = sum_l Ahead[i][l]*s_best[l] + bias[i]
    float Ahead[II][LL], bias[II];
    #pragma unroll
    for (int i = 0; i < II; ++i) {
        bias[i] = 0.0f;
        #pragma unroll
        for (int l = 0; l < LL; ++l) Ahead[i][l] = 0.0f;
        #pragma unroll
        for (int j = 0; j < JJ; ++j) {
            float gsum = g[j][0] + g[j][1];
            bias[i] += gsum * head_b[m*6 + j*3 + i];
            #pragma unroll
            for (int l = 0; l < LL; ++l)
                Ahead[i][l] += head_W[m*12 + j*6 + i*2 + l] * g[j][l];
        }
    }

    // conclusion[l] = sum_l' T[l][l']*s_best[l'] + tb2[l]
    float T[LL][LL], tb2[LL];
    #pragma unroll
    for (int l = 0; l < LL; ++l) {
        tb2[l] = tail_b[m*2 + l];
        #pragma unroll
        for (int lp = 0; lp < LL; ++lp) T[l][lp] = 0.0f;
        #pragma unroll
        for (int i = 0; i < II; ++i) {
            float tw = tail_W[m*6 + l*3 + i];
            tb2[l] += tw * bias[i];
            #pragma unroll
            for (int lp = 0; lp < LL; ++lp)
                T[l][lp] += tw * Ahead[i][lp];
        }
    }

    // ------------------------------------------------------------------
    // Batch loop: one wave processes one batch row b per iteration.
    // ------------------------------------------------------------------
    const int gwave  = (int)((blockIdx.x * blockDim.x + threadIdx.x) >> 5);
    const int nwaves = (int)((gridDim.x * blockDim.x) >> 5);

    for (int b = gwave; b < Bsz; b += nwaves) {
        const float2* sp = reinterpret_cast<const float2*>(state) + (size_t)b * WW;

        // Uniform (wave-wide) load of this row's 18 floats (9 float2, 8B aligned).
        float sv[2 * WW];
        #pragma unroll
        for (int w = 0; w < WW; ++w) {
            float2 t = sp[w];
            sv[2*w]   = t.x;
            sv[2*w+1] = t.y;
        }

        // This lane's WMMA column features: direct lane-indexed load (L0 hit,
        // same cacheline as the uniform load) instead of a cndmask chain.
        float2 fcol = sp[wclamp];
        float fx = colvalid ? fcol.x : 0.0f;
        float fy = colvalid ? fcol.y : 0.0f;
        float fssq = fx*fx + fy*fy;

        // argmin of total_match over the 9 windows (first-min semantics), w=0 peeled
        float bx = sv[0], by = sv[1];
        float mm = K0 + bx * (P[0] + Q[0] * bx) + by * (P[1] + Q[1] * by);
        #pragma unroll
        for (int w = 1; w < WW; ++w) {
            float x = sv[2*w], y = sv[2*w+1];
            float tm = K0 + x * (P[0] + Q[0] * x) + y * (P[1] + Q[1] * y);
            bool take = tm < mm;
            mm = take ? tm : mm;
            bx = take ? x  : bx;
            by = take ? y  : by;
        }

        float c0 = T[0][0]*bx + T[0][1]*by + tb2[0];
        float c1 = T[1][0]*bx + T[1][1]*by + tb2[1];
        // Fold the log2(e) scaling for v_exp_f32 into the A coefficients so the
        // WMMA result feeds v_exp_f32 directly (no post-WMMA pk_mul).
        float a0 = LOG2E * 2.0f * c0;
        float a1 = LOG2E * 2.0f * c1;
        float a2 = LOG2E * (-mm - (c0*c0 + c1*c1));
        float a3 = -LOG2E;

        // D[m,w] = log2e * ( a0'[m]*x_w + a1'[m]*y_w + a2'[m]*1 - (x_w^2+y_w^2) )
        // 16x4 f32 A: lanes<16 hold K=0,1 ; lanes>=16 hold K=2,3 (M = lane&15)
        v2f Amat, Bmat;
        Amat[0] = hi ? a2 : a0;
        Amat[1] = hi ? a3 : a1;
        // 4x16 f32 B: lanes<16 hold K=0,1 ; lanes>=16 hold K=2,3 (N = lane&15)
        Bmat[0] = hi ? fone : fx;
        Bmat[1] = hi ? fssq : fy;
        v8f Cz = {};
        v8f D = __builtin_amdgcn_wmma_f32_16x16x4_f32(
                    false, Amat, false, Bmat, (short)0, Cz, false, false);

        // exp2 + reduce over m: 8 values in-lane cover m=0..7 (lo) / 8..15 (hi);
        // pairing lane <-> lane^16 completes the 16-way m sum for column w.
        float acc = 0.0f;
        #pragma unroll
        for (int i = 0; i < 8; ++i) acc += __builtin_amdgcn_exp2f(D[i]);
        float tot = acc + __shfl_xor(acc, 16, 32);

        if (lane < WW) out[(size_t)b * WW + lane] = tot;
    }
}

extern "C" void kernel_launch(void* const* d_in, const int* in_sizes, int n_in,
                              void* d_out, int out_size, void* d_ws, size_t ws_size,
                              hipStream_t stream) {
    const float* state     = (const float*)d_in[0];
    const float* constants = (const float*)d_in[1];
    const float* gammas    = (const float*)d_in[2];
    const float* head_W    = (const float*)d_in[3];
    const float* head_b    = (const float*)d_in[4];
    const float* tail_W    = (const float*)d_in[5];
    const float* tail_b    = (const float*)d_in[6];
    float* out = (float*)d_out;

    int Bsz = in_sizes[0] / (WW * LL);   // 262144

    const int block = 256;               // 8 waves/block (wave32)
    int blocks = (Bsz + 7) / 8;          // one row per wave-iteration
    if (blocks > 4096) blocks = 4096;    // grid-stride over rows
    if (blocks < 1) blocks = 1;

    algelogic_wmma_kernel<<<blocks, block, 0, stream>>>(
        state, constants, gammas, head_W, head_b, tail_W, tail_b, out, Bsz);
}